// Attention_62715112456978
// MI455X (gfx1250) — compile-verified
//
#include <hip/hip_runtime.h>
#include <math.h>

typedef __attribute__((ext_vector_type(2))) float v2f;
typedef __attribute__((ext_vector_type(8))) float v8f;

#define TB   16    // batch rows per workgroup (WMMA M)
#define NC   576   // concatenated output columns: wk|wq|wv|wo (4*128) + V (64)
#define OSTR 592   // padded LDS row stride (avoid bank conflicts)

__global__ __launch_bounds__(256)
void hyper_attn_kernel(const float* __restrict__ x,
                       const float* __restrict__ state,
                       const float* __restrict__ wk_w, const float* __restrict__ wk_b,
                       const float* __restrict__ wq_w, const float* __restrict__ wq_b,
                       const float* __restrict__ wv_w, const float* __restrict__ wv_b,
                       const float* __restrict__ wo_w, const float* __restrict__ wo_b,
                       const float* __restrict__ V_w,  const float* __restrict__ V_b,
                       float* __restrict__ out)
{
    __shared__ float sOUT[TB * OSTR];   // |wk| |wq| |wv| |wo| v   per batch row
    __shared__ float sX[TB * 64];
    __shared__ float sS1[TB], sS2[TB], sXmax[TB], sXmin[TB];

    const int tid  = threadIdx.x;
    const int wid  = tid >> 5;          // wave id, 0..7 (wave32)
    const int lane = tid & 31;
    const int half = lane >> 4;         // K-half select for A/B operands
    const int ml   = lane & 15;         // M (A) / N (B) index within tile
    const int b0   = blockIdx.x * TB;

    // stage x tile (16 x 64) into LDS
    for (int i = tid; i < TB * 64; i += 256) {
        int m = i >> 6, k = i & 63;
        sX[i] = x[(size_t)(b0 + m) * 64 + k];
    }

    // ---- fp32 WMMA GEMM: OUT = state(16x128) @ Wcat^T + bias, abs() on hypernet cols ----
    for (int n = wid; n < NC / 16; n += 8) {
        const int col0 = n * 16;
        const float* W; const float* Bv; bool do_abs = true;
        if      (col0 < 128) { W = wk_w; Bv = wk_b; }
        else if (col0 < 256) { W = wq_w; Bv = wq_b; }
        else if (col0 < 384) { W = wv_w; Bv = wv_b; }
        else if (col0 < 512) { W = wo_w; Bv = wo_b; }
        else                 { W = V_w;  Bv = V_b; do_abs = false; }
        const int wcol = (col0 < 512) ? (col0 & 127) : (col0 - 512);

        // bias depends only on the output column -> fold into accumulator init
        const float bv = Bv[wcol + ml];
        v8f acc = {bv, bv, bv, bv, bv, bv, bv, bv};

        const float* arow = state + (size_t)(b0 + ml)  * 128 + 2 * half;
        const float* brow = W     + (size_t)(wcol + ml) * 128 + 2 * half;
        for (int k0 = 0; k0 < 128; k0 += 4) {
            v2f a  = *(const v2f*)(arow + k0);   // A: 16x4 fp32 operand
            v2f bb = *(const v2f*)(brow + k0);   // B: 4x16 fp32 operand (W^T)
            acc = __builtin_amdgcn_wmma_f32_16x16x4_f32(false, a, false, bb,
                                                        (short)0, acc, false, false);
        }
        #pragma unroll
        for (int r = 0; r < 8; ++r) {
            float vv = acc[r];                   // row m = r + 8*half, col = col0 + ml
            if (do_abs) vv = fabsf(vv);
            sOUT[(r + 8 * half) * OSTR + col0 + ml] = vv;
        }
    }
    __syncthreads();

    // per-batch x min/max (for a stable softmax max: max_k c*x[k])
    if (tid < TB) {
        float mx = -3.4e38f, mn = 3.4e38f;
        for (int k = 0; k < 64; ++k) {
            float xv = sX[tid * 64 + k];
            mx = fmaxf(mx, xv); mn = fminf(mn, xv);
        }
        sXmax[tid] = mx; sXmin[tid] = mn;
    }

    // s1 = dot(|wq|,|wk|)/sqrt(128), s2 = dot(|wv|,|wo|)  (16 lanes per batch row)
    {
        const int m = tid >> 4, j = tid & 15;
        float p1 = 0.f, p2 = 0.f;
        for (int e = j; e < 128; e += 16) {
            p1 += sOUT[m * OSTR + e]       * sOUT[m * OSTR + 128 + e];
            p2 += sOUT[m * OSTR + 256 + e] * sOUT[m * OSTR + 384 + e];
        }
        #pragma unroll
        for (int off = 8; off >= 1; off >>= 1) {
            p1 += __shfl_xor(p1, off, 16);
            p2 += __shfl_xor(p2, off, 16);
        }
        if (j == 0) { sS1[m] = p1 * 0.08838834764831845f; sS2[m] = p2; }
    }
    __syncthreads();

    // collapsed attention: m[q] = sum_k x[k] softmax_k(s1*x[q]*x[k]); elu head
    for (int p = tid; p < TB * 64; p += 256) {
        const int m = p >> 6, q = p & 63;
        const float c    = sS1[m] * sX[m * 64 + q];
        const float tmax = fmaxf(c * sXmax[m], c * sXmin[m]);
        float se = 0.f, sex = 0.f;
        for (int k = 0; k < 64; ++k) {
            float xk = sX[m * 64 + k];
            float e  = __expf(c * xk - tmax);
            se += e; sex += e * xk;
        }
        const float mval = sex / se;
        const float z = mval * sS2[m] + sOUT[m * OSTR + 512 + q];
        out[(size_t)(b0 + m) * 64 + q] = (z > 0.f) ? z : (__expf(z) - 1.f);
    }
}

extern "C" void kernel_launch(void* const* d_in, const int* in_sizes, int n_in,
                              void* d_out, int out_size, void* d_ws, size_t ws_size,
                              hipStream_t stream) {
    const float* x     = (const float*)d_in[0];
    const float* state = (const float*)d_in[1];
    const float* wk_w  = (const float*)d_in[2];
    const float* wk_b  = (const float*)d_in[3];
    const float* wq_w  = (const float*)d_in[4];
    const float* wq_b  = (const float*)d_in[5];
    const float* wv_w  = (const float*)d_in[6];
    const float* wv_b  = (const float*)d_in[7];
    const float* wo_w  = (const float*)d_in[8];
    const float* wo_b  = (const float*)d_in[9];
    const float* V_w   = (const float*)d_in[10];
    const float* V_b   = (const float*)d_in[11];
    float* out = (float*)d_out;

    const int b = in_sizes[0] / 64;          // x is (b, 1, 64)
    hyper_attn_kernel<<<b / TB, 256, 0, stream>>>(
        x, state, wk_w, wk_b, wq_w, wq_b, wv_w, wv_b, wo_w, wo_b, V_w, V_b, out);
}